// SelfAttention_83124797047010
// MI455X (gfx1250) — compile-verified
//
#include <hip/hip_runtime.h>

typedef float v2f __attribute__((ext_vector_type(2)));
typedef float v8f __attribute__((ext_vector_type(8)));
typedef unsigned int v4u __attribute__((ext_vector_type(4)));
typedef int v4i __attribute__((ext_vector_type(4)));
typedef int v8i __attribute__((ext_vector_type(8)));

#define BATCH 16
#define SEQ   2048
#define DIM   256
#define KPAD  (DIM + 4)      // row pitch 260 f32 = TDM pad: 4 dwords per 256 dwords
#define WAVES 8
#define ROWS_PER_BLOCK 128   // 8 waves x 16 rows

// ---- Tensor Data Mover: load one contiguous 16x256-f32 key tile into LDS ----
// Described as a 1-D 4096-element tile with LDS padding of 4 dwords every 256
// dwords, giving the 260-float row pitch the WMMA fragment reads expect.
__device__ __forceinline__ void tdm_load_tile(const float* gsrc, unsigned lds_byte_addr)
{
    const unsigned long long ga = (unsigned long long)(uintptr_t)gsrc;

    v4u g0;
    g0.x = 1u;                                                // count=1, user descriptor
    g0.y = lds_byte_addr;                                     // lds_addr [63:32]
    g0.z = (unsigned)(ga & 0xffffffffu);                      // global_addr lo
    g0.w = (unsigned)((ga >> 32) & 0x01ffffffu) | (2u << 30); // global_addr hi | type=2

    v8i g1;
    g1[0] = (2 << 16)                    // data_size = 4 bytes
          | (1 << 20)                    // pad_enable
          | (7 << 22)                    // pad_interval: 256 dwords
          | (3 << 25);                   // pad_amount: 4 dwords
    g1[1] = (4096 & 0xffff) << 16;       // tensor_dim0 = 4096 (lo16)
    g1[2] = (4096 >> 16) | (1 << 16);    // tensor_dim0 hi16 | tensor_dim1 = 1
    g1[3] = (4096 << 16);                // tensor_dim1 hi16 | tile_dim0 = 4096
    g1[4] = 1;                           // tile_dim1 = 1 | tile_dim2 = 0
    g1[5] = 4096;                        // tensor_dim0_stride lo32
    g1[6] = 0;                           // stride hi | tensor_dim1_stride lo
    g1[7] = 0;

    v4i g2 = (v4i){1, 1, 0, 0};          // tensor_dim2=1, tensor_dim3=1 (unused, 1-D tile)
    v4i g3 = (v4i){0, 1 << 16, 0, 0};    // tensor_dim4=1 (unused)

#if defined(__clang_major__) && (__clang_major__ >= 23)
    __builtin_amdgcn_tensor_load_to_lds(g0, g1, g2, g3, (v8i){0,0,0,0,0,0,0,0}, 0);
#else
    __builtin_amdgcn_tensor_load_to_lds(g0, g1, g2, g3, 0);
#endif
}

// Flash-attention style fused self-attention, fp32 WMMA (V_WMMA_F32_16X16X4_F32),
// TDM-staged, double-buffered K/V tiles (K-tile == V-tile for self attention).
__global__ __launch_bounds__(256, 1)
void SelfAttention_fp32_wmma(const float* __restrict__ x,
                             const int*   __restrict__ length,
                             float*       __restrict__ out)
{
    __shared__ __align__(16) float kt[2][16][KPAD];     // double-buffered 16x256 tiles
    __shared__ __align__(16) float pl[WAVES][16][20];   // per-wave P tile scratch (padded)

    const int tid  = threadIdx.x;
    const int lane = tid & 31;
    const int wv   = tid >> 5;
    const int l15  = lane & 15;
    const int hl   = lane >> 4;   // lane half: selects K-pair / row-group per WMMA layout

    const int b    = blockIdx.x >> 4;   // 16 row-blocks per batch
    const int rb   = blockIdx.x & 15;
    const int row0 = rb * ROWS_PER_BLOCK + wv * 16;

    const int len    = length[b];
    const int ntiles = (len + 15) >> 4;   // tiles with j0 >= len contribute exp(-inf)=0: skip

    const float* xb = x + (size_t)b * SEQ * DIM;

    // ---- Q fragments in registers: A-matrix layout for 16x16x4 f32 WMMA ----
    // lane l holds row l%16, dims {4s + 2*(l/16), +1}
    const float* qbase = xb + (size_t)(row0 + l15) * DIM + 2 * hl;
    v2f qf[64];
#pragma unroll
    for (int s = 0; s < 64; ++s)
        qf[s] = *(const v2f*)(qbase + 4 * s);

    v8f acc[16];   // 16 output d-tiles, C/D layout: reg r -> row r + 8*hl, col = 16t + l15
#pragma unroll
    for (int t = 0; t < 16; ++t)
        acc[t] = (v8f){0.f, 0.f, 0.f, 0.f, 0.f, 0.f, 0.f, 0.f};

    float mrow[8], lrow[8];    // per-row running max / running sum (rows r + 8*hl)
#pragma unroll
    for (int r = 0; r < 8; ++r) { mrow[r] = -__builtin_inff(); lrow[r] = 0.f; }

    // ---- prologue: TDM-stage tile 0 into buffer 0 ----
    if (wv == 0) {
        tdm_load_tile(xb, (unsigned)(uintptr_t)&kt[0][0][0]);
        __builtin_amdgcn_s_wait_tensorcnt(0);
    }
    __syncthreads();

    for (int kti = 0; kti < ntiles; ++kti) {
        const int j0  = kti * 16;
        const int cur = kti & 1;

        // ---- async TDM prefetch of the next tile into the other buffer ----
        if (wv == 0 && (kti + 1) < ntiles)
            tdm_load_tile(xb + (size_t)(j0 + 16) * DIM,
                          (unsigned)(uintptr_t)&kt[cur ^ 1][0][0]);

        const float (*K)[KPAD] = kt[cur];

        // ---- S = Q * K^T  (16 queries x 16 keys), K-dim 256 in 64 WMMA steps ----
        // Software-pipelined: 4 B-fragments (ds_load_b64) kept in flight so each
        // load has ~4 WMMAs of slack instead of s_wait_dscnt 0 on the consumer.
        v2f bf[4];
#pragma unroll
        for (int i = 0; i < 4; ++i)
            bf[i] = *(const v2f*)&K[l15][4 * i + 2 * hl];

        v8f S = (v8f){0.f, 0.f, 0.f, 0.f, 0.f, 0.f, 0.f, 0.f};
#pragma unroll
        for (int s = 0; s < 64; ++s) {
            // B-matrix layout: lane l holds key n = l%16, dims {4s + 2*(l/16), +1}
            const v2f curf = bf[s & 3];
            if (s + 4 < 64)
                bf[s & 3] = *(const v2f*)&K[l15][4 * (s + 4) + 2 * hl];
            S = __builtin_amdgcn_wmma_f32_16x16x4_f32(false, qf[s], false, curf,
                                                      (short)0, S, false, false);
        }

        // ---- masked online softmax (column mask: j < length[b]) ----
        const bool valid = (j0 + l15) < len;
        float scl[8];
#pragma unroll
        for (int r = 0; r < 8; ++r) {
            float sv = valid ? S[r] : -__builtin_inff();
            // row max across the 16 lanes holding this row (xor<16 stays in-half)
            float tm = sv;
#pragma unroll
            for (int m = 8; m >= 1; m >>= 1)
                tm = fmaxf(tm, __shfl_xor(tm, m, 32));
            const float mn = fmaxf(mrow[r], tm);        // finite: j0 < len => a valid col exists
            const float sc = __expf(mrow[r] - mn);      // first tile: exp(-inf)=0
            const float p  = __expf(sv - mn);           // masked: exp(-inf)=0
            float rs = p;
#pragma unroll
            for (int m = 8; m >= 1; m >>= 1)
                rs += __shfl_xor(rs, m, 32);
            lrow[r] = lrow[r] * sc + rs;
            mrow[r] = mn;
            scl[r]  = sc;
            // D-layout -> LDS so we can re-read P in A-layout for the PV matmul
            pl[wv][r + 8 * hl][l15] = p;
        }

        v8f sfac;
#pragma unroll
        for (int r = 0; r < 8; ++r) sfac[r] = scl[r];
#pragma unroll
        for (int t = 0; t < 16; ++t) acc[t] *= sfac;

        // ---- acc += P * V  (V == staged key tile), 16 d-tiles x 4 k-steps ----
        v2f af[4];
#pragma unroll
        for (int kk = 0; kk < 4; ++kk)   // A frag: lane l holds row l%16, keys {4kk+2*hl, +1}
            af[kk] = *(const v2f*)&pl[wv][l15][4 * kk + 2 * hl];

        // Software-pipelined over d-tiles: prefetch B-fragments for tile t+1
        // while the 4 WMMAs of tile t execute.
        v2f vb[4], vbn[4];
#pragma unroll
        for (int kk = 0; kk < 4; ++kk) {
            const int kr = 4 * kk + 2 * hl;
            vb[kk].x = K[kr    ][l15];
            vb[kk].y = K[kr + 1][l15];
        }
#pragma unroll
        for (int t = 0; t < 16; ++t) {
            if (t + 1 < 16) {
#pragma unroll
                for (int kk = 0; kk < 4; ++kk) {
                    const int kr = 4 * kk + 2 * hl;
                    vbn[kk].x = K[kr    ][16 * (t + 1) + l15];
                    vbn[kk].y = K[kr + 1][16 * (t + 1) + l15];
                }
            }
            v8f a = acc[t];
#pragma unroll
            for (int kk = 0; kk < 4; ++kk)
                a = __builtin_amdgcn_wmma_f32_16x16x4_f32(false, af[kk], false, vb[kk],
                                                          (short)0, a, false, false);
            acc[t] = a;
#pragma unroll
            for (int kk = 0; kk < 4; ++kk)
                vb[kk] = vbn[kk];
        }

        // ---- publish the prefetched buffer for the next iteration ----
        if (wv == 0)
            __builtin_amdgcn_s_wait_tensorcnt(0);
        __syncthreads();
    }

    // ---- normalize and store ----
    float inv[8];
#pragma unroll
    for (int r = 0; r < 8; ++r) inv[r] = 1.f / lrow[r];   // len>=1 => lrow>0

    float* obase = out + ((size_t)b * SEQ + row0 + 8 * hl) * DIM + l15;
#pragma unroll
    for (int t = 0; t < 16; ++t)
#pragma unroll
        for (int r = 0; r < 8; ++r)
            obase[(size_t)r * DIM + 16 * t] = acc[t][r] * inv[r];
}

extern "C" void kernel_launch(void* const* d_in, const int* in_sizes, int n_in,
                              void* d_out, int out_size, void* d_ws, size_t ws_size,
                              hipStream_t stream) {
    (void)in_sizes; (void)n_in; (void)out_size; (void)d_ws; (void)ws_size;
    const float* x      = (const float*)d_in[0];
    const int*   length = (const int*)d_in[1];
    float*       out    = (float*)d_out;

    const dim3 grid(BATCH * (SEQ / ROWS_PER_BLOCK));   // 256 blocks
    const dim3 block(256);                              // 8 wave32 waves
    SelfAttention_fp32_wmma<<<grid, block, 0, stream>>>(x, length, out);
}